// SpectraIntegratorStationary_28724741275993
// MI455X (gfx1250) — compile-verified
//
#include <hip/hip_runtime.h>
#include <hip/hip_bf16.h>
#include <math.h>

// CDNA5 / gfx1250, wave32.
typedef __attribute__((ext_vector_type(2))) float v2f;
typedef __attribute__((ext_vector_type(8))) float v8f;

#define TILE_M 128   // lines staged in LDS per stage (2KB)
#define NT     128   // field points per block (8 waves x 16 rows)
#define MSPLIT 8     // m-chunks -> blocks, combined via f32 atomics

__global__ void zero_out_kernel(float* __restrict__ out, int n) {
    int i = blockIdx.x * blockDim.x + threadIdx.x;
    if (i < n) out[i] = 0.0f;
}

// Per-(b,m) hoisted math. Sort is dropped: B_mean and the pairwise squared
// differences are symmetric in (B1,B2,B3).
__global__ void precompute_kernel(const float* __restrict__ rf,     // [B,M,3]
                                  const float* __restrict__ width,  // [B,M]
                                  const float* __restrict__ Am,     // [B,M]
                                  const float* __restrict__ area,   // [M]
                                  const float* __restrict__ F,      // [B,N]
                                  float4* __restrict__ params,      // [B,M]
                                  int B, int M, int N) {
    int i = blockIdx.x * blockDim.x + threadIdx.x;
    if (i >= B * M) return;
    int b = i / M;
    int m = i - b * M;

    const float nw = 1e-5f;
    float sw = (F[(size_t)b * N + 1] - F[(size_t)b * N]) * 0.5f;

    float A = Am[i] * area[m];
    float w = width[i];
    w = (w > nw) ? w : w + nw;

    float r1 = rf[3 * (size_t)i + 0];
    float r2 = rf[3 * (size_t)i + 1];
    float r3 = rf[3 * (size_t)i + 2];
    float d12 = r1 - r2, d13 = r1 - r3, d23 = r2 - r3;
    float inv_w = 1.0f / w;
    float add_w_sq = (d13 * d13 + d23 * d23 + d12 * d12) * (inv_w * inv_w) * (1.0f / 9.0f);

    float w2 = (w > 2.0f * nw) ? w : w + 2.0f * nw;
    float ew = sqrtf(w2 * w2 * (1.0f + add_w_sq) + sw * sw);
    ew = (ew < 0.5f * sw) ? ew : ew + 0.5f * sw;

    float Bmean = (r1 + r2 + r3) * (1.0f / 3.0f);
    float c = 1.41421356237309504880f / ew;                 // sqrt(2)/ew
    float coef = A * 2.0f * c * c * 0.5641895835477562869f; // 2*A*c^2/sqrt(pi)
    params[i] = make_float4(Bmean, c, coef, 0.0f);
}

__device__ __forceinline__ float lineval(float4 p, float f) {
    float arg = (p.x - f) * p.y;              // (B_mean - field) * c
    float e = __expf(-arg * arg);             // v_exp_f32 (TRANS)
    float v = p.z * arg * e;                  // coef * arg * exp(-arg^2)
    return (fabsf(arg) <= 3.0f) ? v : 0.0f;   // v_cndmask, no EXEC change
}

// Each wave owns 16 consecutive n (A-matrix rows). Per step, 4 lines of the
// LDS tile feed one V_WMMA_F32_16X16X4_F32 with an all-ones B operand:
// D[row, *] += sum_k lineval(row, m0+k), i.e. the m-reduction rides the
// matrix pipe while VALU/TRANS compute the next line shapes.
__global__ void spectra_wmma_kernel(const float4* __restrict__ params, // [B,M]
                                    const float* __restrict__ F,       // [B,N]
                                    float* __restrict__ out,           // [B,N]
                                    int M, int N, int chunk) {
    __shared__ float4 smem[TILE_M];

    const int tid  = threadIdx.x;
    const int lane = tid & 31;              // wave32
    const int wave = tid >> 5;
    const int b    = blockIdx.z;

    const int nBase = blockIdx.x * NT + wave * 16;
    const int row   = lane & 15;            // A-matrix row this lane feeds
    const int kSel  = lane >> 4;            // lanes 0-15 -> K=0,1; 16-31 -> K=2,3
    const int n     = nBase + row;
    const float f   = (n < N) ? F[(size_t)b * N + n] : 0.0f;

    const int mStart = blockIdx.y * chunk;
    const int mEnd   = (mStart + chunk < M) ? (mStart + chunk) : M;
    const float4* pbase = params + (size_t)b * M;

    v8f acc = {0.0f, 0.0f, 0.0f, 0.0f, 0.0f, 0.0f, 0.0f, 0.0f};
    v2f ones;
    ones.x = 1.0f;
    ones.y = 1.0f;

    for (int mt = mStart; mt < mEnd; mt += TILE_M) {
        int cnt = mEnd - mt;
        if (cnt > TILE_M) cnt = TILE_M;

        __syncthreads();
        for (int i = tid; i < TILE_M; i += blockDim.x) {
            // coef=0, c=0 pad -> lineval returns exactly 0 for tail entries
            smem[i] = (i < cnt) ? pbase[mt + i] : make_float4(0.f, 0.f, 0.f, 0.f);
        }
        __syncthreads();

        if (mt + TILE_M < mEnd && tid < 32) {   // hint next tile into cache
            __builtin_prefetch((const void*)(pbase + mt + TILE_M + lane * 4), 0, 1);
        }

#pragma unroll 4
        for (int s = 0; s < TILE_M; s += 4) {
            float4 p0 = smem[s + 2 * kSel];     // broadcast reads, bank-clean
            float4 p1 = smem[s + 2 * kSel + 1];
            v2f av;
            av.x = lineval(p0, f);
            av.y = lineval(p1, f);
            // (neg_a, A, neg_b, B, c_mod, C, reuse_a, reuse_b)
            acc = __builtin_amdgcn_wmma_f32_16x16x4_f32(
                false, av, false, ones, (short)0, acc, false, false);
        }
    }

    // D layout: VGPR r, lanes 0-15 hold row r; lanes 16-31 hold row r+8,
    // replicated across all 16 columns (B was all-ones). Lane 0 writes rows
    // 0..7, lane 16 writes rows 8..15.
    if ((lane & 15) == 0) {
        int base = nBase + kSel * 8;
#pragma unroll
        for (int r = 0; r < 8; ++r) {
            int nn = base + r;
            if (nn < N) atomicAdd(&out[(size_t)b * N + nn], acc[r]);
        }
    }
}

extern "C" void kernel_launch(void* const* d_in, const int* in_sizes, int n_in,
                              void* d_out, int out_size, void* d_ws, size_t ws_size,
                              hipStream_t stream) {
    const float* res_fields = (const float*)d_in[0]; // [B,M,3]
    const float* width      = (const float*)d_in[1]; // [B,M]
    const float* A_mean     = (const float*)d_in[2]; // [B,M]
    const float* area       = (const float*)d_in[3]; // [M]
    const float* F          = (const float*)d_in[4]; // [B,N]
    float* out = (float*)d_out;

    const int M = in_sizes[3];
    const int B = in_sizes[1] / M;
    const int N = in_sizes[4] / B;

    float4* params = (float4*)d_ws; // needs B*M*16 bytes (256 KB here)

    zero_out_kernel<<<(out_size + 255) / 256, 256, 0, stream>>>(out, out_size);

    precompute_kernel<<<(B * M + 255) / 256, 256, 0, stream>>>(
        res_fields, width, A_mean, area, F, params, B, M, N);

    const int chunk = (M + MSPLIT - 1) / MSPLIT;
    dim3 grid((N + NT - 1) / NT, MSPLIT, B);
    spectra_wmma_kernel<<<grid, 256, 0, stream>>>(params, F, out, M, N, chunk);
}